// loss_mod_Memory_25177098289566
// MI455X (gfx1250) — compile-verified
//
#include <hip/hip_runtime.h>
#include <hip/hip_bf16.h>
#include <math.h>

#define BB 64
#define CC 64
#define SS 128
#define HH 512
#define DD (SS * HH)
#define EPSF 1e-8f
#define NSPLIT 16

typedef __attribute__((ext_vector_type(2))) float v2f;
typedef __attribute__((ext_vector_type(8))) float v8f;
typedef __attribute__((ext_vector_type(4))) int   v4i;

#if defined(__AMDGCN__) && \
    __has_builtin(__builtin_amdgcn_global_load_async_to_lds_b128) && \
    __has_builtin(__builtin_amdgcn_s_wait_asynccnt)
#define HAVE_ASYNC_LDS 1
#else
#define HAVE_ASYNC_LDS 0
#endif

// ---------------------------------------------------------------------------
// Row squared norms: out[r] = sum_k A[r,k]^2
// ---------------------------------------------------------------------------
__global__ void rownorm2_kernel(const float* __restrict__ A, float* __restrict__ out, int K) {
    __shared__ float red[256];
    const int r = blockIdx.x;
    const float* row = A + (size_t)r * K;
    float s = 0.0f;
    const int K4 = K >> 2;
    for (int k = threadIdx.x; k < K4; k += 256) {
        float4 v = ((const float4*)row)[k];
        s += v.x * v.x + v.y * v.y + v.z * v.z + v.w * v.w;
    }
    red[threadIdx.x] = s;
    __syncthreads();
    for (int off = 128; off > 0; off >>= 1) {
        if (threadIdx.x < off) red[threadIdx.x] += red[threadIdx.x + off];
        __syncthreads();
    }
    if (threadIdx.x == 0) out[r] = red[0];
}

// ---------------------------------------------------------------------------
// K-split 64x64 NT-GEMM partials via V_WMMA_F32_16X16X4_F32:
//   part[split][m,n] = sum_{k in split chunk} A[m,k] * Bm[n,k]
// grid = 16*NSPLIT blocks, 32 threads (1 wave) each; block = (tile, K-chunk).
// fp32 A-frag (16x4): lane<16 -> row m0+lane, K={k,k+1}; lane>=16 -> K={k+2,k+3}
// fp32 B-frag (4x16): identical lane mapping on Bm rows (NT symmetry).
// ---------------------------------------------------------------------------
__global__ void wmma_nt_gemm64_part_kernel(const float* __restrict__ A,
                                           const float* __restrict__ Bm,
                                           float* __restrict__ part, int kchunk) {
    const int tile  = blockIdx.x & 15;
    const int split = blockIdx.x >> 4;
    const int m0 = (tile >> 2) * 16;
    const int n0 = (tile & 3) * 16;
    const int lane = threadIdx.x & 31;
    const int l = lane & 15;
    const int hi = lane >> 4;
    const size_t K = (size_t)kchunk * NSPLIT;
    const int kbeg = split * kchunk;

    const float* ar = A  + (size_t)(m0 + l) * K + 2 * hi + kbeg;
    const float* br = Bm + (size_t)(n0 + l) * K + 2 * hi + kbeg;

    v8f acc = {};
    for (int k = 0; k < kchunk; k += 4) {
        v2f a = *(const v2f*)(ar + k);
        v2f b = *(const v2f*)(br + k);
        acc = __builtin_amdgcn_wmma_f32_16x16x4_f32(
            false, a, false, b, (short)0, acc, false, false);
    }
    float* po = part + (size_t)split * 4096;
#pragma unroll
    for (int v = 0; v < 8; ++v) {
        int row = m0 + v + 8 * hi;           // C/D layout: VGPR v -> M=v (+8 for hi lanes)
        po[row * 64 + n0 + l] = acc[v];
    }
}

// Fixed-order (deterministic) reduction of NSPLIT partials -> out[4096]
__global__ void reduce_part_kernel(const float* __restrict__ part, float* __restrict__ out) {
    const int i = blockIdx.x * 256 + threadIdx.x;   // 0..4095
    float s = 0.0f;
    for (int p = 0; p < NSPLIT; ++p) s += part[(size_t)p * 4096 + i];
    out[i] = s;
}

// ---------------------------------------------------------------------------
// Batched Gram: G[b] = X_b (128x512) @ X_b^T  -> 128x128
// One workgroup (256 thr = 8 waves) per batch element. K staged via LDS in
// 128x64 chunks (34KB, pad 68). Wave w owns the 8 tiles of output column-block
// tn = w (tm = 0..7), so the B-frag is shared by all 8 WMMAs per k-step:
// 9 LDS b64 reads feed 8 WMMAs into 8 independent accumulators (no RAW NOPs).
// Staging uses GLOBAL_LOAD_ASYNC_TO_LDS_B128 (ASYNCcnt) when available.
// ---------------------------------------------------------------------------
#define KCHUNK 64
#define LPAD   68   // mult of 4 (b128 stores), even (b64 reads), 68%64 -> bank-skewed

__global__ void __launch_bounds__(256) gram_wmma_kernel(const float* __restrict__ Xall,
                                                        float* __restrict__ Gall) {
    __shared__ float sX[SS * LPAD];
    const int b = blockIdx.x;
    const float* X = Xall + (size_t)b * SS * HH;
    float* G = Gall + (size_t)b * SS * SS;

    const int tid  = threadIdx.x;
    const int lane = tid & 31;
    const int wave = tid >> 5;        // 0..7  == tn column-block
    const int l  = lane & 15;
    const int hi = lane >> 4;

    v8f acc[8] = {};
    const float* brow = &sX[(wave * 16 + l) * LPAD + 2 * hi];

    for (int k0 = 0; k0 < HH; k0 += KCHUNK) {
        __syncthreads();   // previous chunk fully consumed before overwrite
#if HAVE_ASYNC_LDS
        for (int i = tid; i < SS * (KCHUNK / 4); i += 256) {
            int row = i >> 4;
            int c4  = i & 15;
            __builtin_amdgcn_global_load_async_to_lds_b128(
                (v4i*)(X + (size_t)row * HH + k0 + c4 * 4),
                (v4i*)&sX[row * LPAD + c4 * 4], 0, 0);
        }
        __builtin_amdgcn_s_wait_asynccnt(0);
#else
        for (int i = tid; i < SS * (KCHUNK / 4); i += 256) {
            int row = i >> 4;
            int c4  = i & 15;
            float4 v = *(const float4*)(X + (size_t)row * HH + k0 + c4 * 4);
            *(float4*)(&sX[row * LPAD + c4 * 4]) = v;
        }
#endif
        __syncthreads();
        for (int kk = 0; kk < KCHUNK; kk += 4) {
            v2f bb = *(const v2f*)(brow + kk);      // shared across the 8 tiles
#pragma unroll
            for (int t = 0; t < 8; ++t) {           // tm = t
                v2f a = *(const v2f*)(&sX[(t * 16 + l) * LPAD + 2 * hi + kk]);
                acc[t] = __builtin_amdgcn_wmma_f32_16x16x4_f32(
                    false, a, false, bb, (short)0, acc[t], false, false);
            }
        }
    }
#pragma unroll
    for (int t = 0; t < 8; ++t) {                   // tm = t, tn = wave
#pragma unroll
        for (int v = 0; v < 8; ++v) {
            int row = t * 16 + v + 8 * hi;
            G[row * SS + wave * 16 + l] = acc[t][v];
        }
    }
}

// ---------------------------------------------------------------------------
// In-place centering of symmetric Gram + h = sum(cg^2).
// cg = g - rowmean - colmean + grandmean ; symmetry => colmean == rowmean.
// ---------------------------------------------------------------------------
__global__ void center_kernel(float* __restrict__ Gall, float* __restrict__ hout) {
    __shared__ float rm[SS];
    __shared__ float red[SS];
    const int b = blockIdx.x;
    float* G = Gall + (size_t)b * SS * SS;
    const int t = threadIdx.x;    // 0..127

    float s = 0.0f;
    for (int c = 0; c < SS; ++c) s += G[t * SS + c];
    rm[t] = s * (1.0f / SS);
    __syncthreads();

    float g = 0.0f;
    for (int c = 0; c < SS; ++c) g += rm[c];
    const float gm = g * (1.0f / SS);

    const float mrow = rm[t];
    float sq = 0.0f;
    for (int c = 0; c < SS; ++c) {
        float cg = G[t * SS + c] - mrow - rm[c] + gm;
        G[t * SS + c] = cg;
        sq += cg * cg;
    }
    red[t] = sq;
    __syncthreads();
    for (int off = 64; off > 0; off >>= 1) {
        if (t < off) red[t] += red[t + off];
        __syncthreads();
    }
    if (t == 0) hout[b] = red[0];
}

// ---------------------------------------------------------------------------
// dist[b,c] = sqrt(max(x2[b] + w2[c] - 2*dot[b,c], 0))
// ---------------------------------------------------------------------------
__global__ void dist_kernel(const float* __restrict__ dot, const float* __restrict__ x2,
                            const float* __restrict__ w2, float* __restrict__ dist_out) {
    const int i = blockIdx.x * blockDim.x + threadIdx.x;   // 0..4095
    const int b = i >> 6, c = i & 63;
    float d2 = x2[b] + w2[c] - 2.0f * dot[i];
    dist_out[i] = sqrtf(fmaxf(d2, 0.0f));
}

// ---------------------------------------------------------------------------
// idx[b] = argmax_c  -log(|hxy|/(sqrt(hxx)sqrt(hyy)) + eps)   (first max kept)
// ---------------------------------------------------------------------------
__global__ void argmax_kernel(const float* __restrict__ hxy, const float* __restrict__ hxx,
                              const float* __restrict__ hyy, int* __restrict__ idx_ws,
                              float* __restrict__ idx_out) {
    const int b = threadIdx.x;    // 64 threads
    const float ix = 1.0f / sqrtf(hxx[b]);
    float best = -3.4e38f;
    int bi = 0;
    for (int c = 0; c < 64; ++c) {
        float ratio = fabsf(hxy[b * 64 + c]) * ix / sqrtf(hyy[c]);
        float cka = -logf(ratio + EPSF);
        if (cka > best) { best = cka; bi = c; }
    }
    idx_ws[b] = bi;
    idx_out[b] = (float)bi;
}

// ---------------------------------------------------------------------------
// cka_loss = -log(mean_{b,c} |hxy[b,idx[c]]| / (sqrt(hxx[b]) sqrt(hyy[idx[c]])) + eps)
// (cgs[c] == cgy[idx[c]], so hxy2 is a gather of hxy)
// ---------------------------------------------------------------------------
__global__ void loss_kernel(const float* __restrict__ hxy, const float* __restrict__ hxx,
                            const float* __restrict__ hyy, const int* __restrict__ idx,
                            float* __restrict__ loss_out) {
    __shared__ float red[256];
    const int t = threadIdx.x;
    float s = 0.0f;
    for (int i = t; i < 4096; i += 256) {
        int b = i >> 6, c = i & 63;
        int j = idx[c];
        s += fabsf(hxy[b * 64 + j]) / (sqrtf(hxx[b]) * sqrtf(hyy[j]));
    }
    red[t] = s;
    __syncthreads();
    for (int off = 128; off > 0; off >>= 1) {
        if (t < off) red[t] += red[t + off];
        __syncthreads();
    }
    if (t == 0) loss_out[0] = -logf(red[0] * (1.0f / 4096.0f) + EPSF);
}

// ---------------------------------------------------------------------------
extern "C" void kernel_launch(void* const* d_in, const int* in_sizes, int n_in,
                              void* d_out, int out_size, void* d_ws, size_t ws_size,
                              hipStream_t stream) {
    const float* x = (const float*)d_in[0];   // (B, D)
    const float* w = (const float*)d_in[1];   // (C, D)

    float* out      = (float*)d_out;
    float* loss_out = out;                                  // 1
    float* dist_out = out + 1;                              // 4096
    float* cw_out   = out + 1 + 4096;                       // C*D
    float* idx_out  = out + 1 + 4096 + (size_t)CC * DD;     // 64

    float* ws   = (float*)d_ws;
    float* x2   = ws;                 // 64
    float* w2   = ws + 64;            // 64
    float* dot  = ws + 128;           // 4096
    float* hxy  = ws + 128 + 4096;    // 4096
    float* hxx  = hxy + 4096;         // 64
    float* hyy  = hxx + 64;           // 64
    int*   idxw = (int*)(hyy + 64);   // 64
    float* part = ws + 65536;                  // NSPLIT*4096 = 65536
    float* cgx  = ws + 131072;                 // B*S*S = 1,048,576
    float* cgy  = cgx + (size_t)BB * SS * SS;  // C*S*S = 1,048,576

    // pass-through output
    (void)hipMemcpyAsync(cw_out, w, sizeof(float) * (size_t)CC * DD,
                         hipMemcpyDeviceToDevice, stream);

    // cdist pieces: x.w^T split 16 ways over K, then deterministic reduce
    rownorm2_kernel<<<BB, 256, 0, stream>>>(x, x2, DD);
    rownorm2_kernel<<<CC, 256, 0, stream>>>(w, w2, DD);
    wmma_nt_gemm64_part_kernel<<<16 * NSPLIT, 32, 0, stream>>>(x, w, part, DD / NSPLIT);
    reduce_part_kernel<<<16, 256, 0, stream>>>(part, dot);
    dist_kernel<<<16, 256, 0, stream>>>(dot, x2, w2, dist_out);

    // centered Grams + self-HSIC
    gram_wmma_kernel<<<BB, 256, 0, stream>>>(x, cgx);
    gram_wmma_kernel<<<CC, 256, 0, stream>>>(w, cgy);
    center_kernel<<<BB, 128, 0, stream>>>(cgx, hxx);
    center_kernel<<<CC, 128, 0, stream>>>(cgy, hyy);

    // cross-HSIC: 64x64 NT-GEMM over flattened centered grams (K = S*S)
    wmma_nt_gemm64_part_kernel<<<16 * NSPLIT, 32, 0, stream>>>(cgx, cgy, part,
                                                               (SS * SS) / NSPLIT);
    reduce_part_kernel<<<16, 256, 0, stream>>>(part, hxy);

    // selection + loss
    argmax_kernel<<<1, 64, 0, stream>>>(hxy, hxx, hyy, idxw, idx_out);
    loss_kernel<<<1, 256, 0, stream>>>(hxy, hxx, hyy, idxw, loss_out);
}